// TemporalGAT_46093589020834
// MI455X (gfx1250) — compile-verified
//
#include <hip/hip_runtime.h>
#include <math.h>

// ---- problem constants (match reference) ----
#define Nn 50000
#define Ee 500000
#define FIN 15
#define Hh 8
#define Cc 16
#define HC 128          // Hh*Cc
#define Gg 64
#define HID 16
#define NCLS 7
#define NEG_SLOPE 0.2f
#define BN_EPS 1e-5f

typedef float v2f __attribute__((ext_vector_type(2)));
typedef float v8f __attribute__((ext_vector_type(8)));

__device__ __forceinline__ float lrelu(float x) { return x > 0.f ? x : NEG_SLOPE * x; }
__device__ __forceinline__ float sigmoidf(float x) { return 1.f / (1.f + expf(-x)); }

// float atomic max via int/uint punning (works for mixed signs, init = -inf)
__device__ __forceinline__ void atomicMaxF(float* addr, float v) {
  if (v >= 0.f) atomicMax((int*)addr, __float_as_int(v));
  else          atomicMin((unsigned int*)addr, __float_as_uint(v));
}

// ---------------------------------------------------------------------------
// Y[Nn,HC] = X[Nn,K] @ W[K,HC] + bias
// One wave32 per (16-row x 64-col) strip: 4 accumulator tiles share one A frag.
// V_WMMA_F32_16X16X4_F32 layouts (32-bit operands):
//   A 16x4: lanes 0-15 row M=l  (v0=K+0, v1=K+1); lanes 16-31 row M=l-16 (K+2,K+3)
//   B 4x16: lanes 0-15 col N=l  (v0=K+0, v1=K+1); lanes 16-31 col N=l-16 (K+2,K+3)
//   D 16x16: VGPR j -> M=j (lanes 0-15) / M=j+8 (lanes 16-31), N = lane&15
// K is a template parameter so tail guards are branch-free (clamped index +
// cndmask select) and K%4==0 uses straight b64 loads for A.
// ---------------------------------------------------------------------------
template <int K>
__global__ __launch_bounds__(32) void gemm_wmma(const float* __restrict__ X,
                                                const float* __restrict__ W,
                                                const float* __restrict__ bias,
                                                float* __restrict__ Y) {
  const int lane = threadIdx.x & 31;
  const int half = lane >> 4;
  const int l16  = lane & 15;
  const int row0 = blockIdx.x * 16;
  const int col0 = blockIdx.y * 64;        // 4 tiles of 16 columns
  const float* Xrow = X + (size_t)(row0 + l16) * K;

  v8f zero = {};
  v8f acc[4];
#pragma unroll
  for (int t = 0; t < 4; ++t) acc[t] = zero;

#pragma unroll 4
  for (int k = 0; k < K; k += 4) {
    const int ka = k + 2 * half;           // this lane-half's K pair: {ka, ka+1}
    v2f a;
    if constexpr ((K & 3) == 0) {
      a = *(const v2f*)(Xrow + ka);        // aligned b64 load
    } else {
      const bool ok1 = (ka + 1 < K);       // ka itself always < K for K=15 path
      const int ka1  = ok1 ? (ka + 1) : (K - 1);
      a.x = Xrow[ka];
      float ay = Xrow[ka1];
      a.y = ok1 ? ay : 0.f;
    }
    const bool ok1 = ((K & 3) == 0) || (ka + 1 < K);
    const int kb0 = ka;
    const int kb1 = ok1 ? (ka + 1) : (K - 1);
    const float* Wr0 = W + kb0 * HC + col0 + l16;
    const float* Wr1 = W + kb1 * HC + col0 + l16;
#pragma unroll
    for (int t = 0; t < 4; ++t) {
      v2f b;
      b.x = Wr0[t * 16];
      float by = Wr1[t * 16];
      b.y = ok1 ? by : 0.f;
      acc[t] = __builtin_amdgcn_wmma_f32_16x16x4_f32(
          /*neg_a=*/false, a, /*neg_b=*/false, b,
          /*c_mod=*/(short)0, acc[t], /*reuse_a=*/false, /*reuse_b=*/false);
    }
  }

#pragma unroll
  for (int t = 0; t < 4; ++t) {
    const int col = col0 + t * 16 + l16;
    const float bv = bias[col];
#pragma unroll
    for (int j = 0; j < 8; ++j)
      Y[(row0 + j + 8 * half) * HC + col] = acc[t][j] + bv;
  }
}

__global__ void fill_f32(float* __restrict__ p, float v, int n) {
  int t = blockIdx.x * blockDim.x + threadIdx.x;
  if (t < n) p[t] = v;
}

// one wave per edge: e[e,h] = att_h . leakyrelu(xl[src] + xr[dst]); atomic segment max
__global__ void edge_scores(const float* __restrict__ XL, const float* __restrict__ XR,
                            const int* __restrict__ src, const int* __restrict__ dst,
                            const float* __restrict__ att, float* __restrict__ esc,
                            float* __restrict__ emax) {
  int e = (blockIdx.x * blockDim.x + threadIdx.x) >> 5;
  if (e >= Ee) return;
  int lane = threadIdx.x & 31;
  int s = src[e], d = dst[e];
  float4 xl = *(const float4*)(XL + s * HC + lane * 4);
  float4 xr = *(const float4*)(XR + d * HC + lane * 4);
  float4 at = *(const float4*)(att + lane * 4);
  float p = lrelu(xl.x + xr.x) * at.x + lrelu(xl.y + xr.y) * at.y +
            lrelu(xl.z + xr.z) * at.z + lrelu(xl.w + xr.w) * at.w;
  p += __shfl_xor(p, 1, 32);
  p += __shfl_xor(p, 2, 32);
  if ((lane & 3) == 0) {
    int h = lane >> 2;
    esc[e * Hh + h] = p;
    atomicMaxF(&emax[d * Hh + h], p);
  }
}

// w = exp(e - emax[dst]); denom[dst] += w   (one thread per (edge,head))
__global__ void edge_exp(const int* __restrict__ dst, float* __restrict__ esc,
                         const float* __restrict__ emax, float* __restrict__ den) {
  int t = blockIdx.x * blockDim.x + threadIdx.x;
  if (t >= Ee * Hh) return;
  int e = t >> 3, h = t & 7;
  int d = dst[e];
  float w = expf(esc[t] - emax[d * Hh + h]);
  esc[t] = w;
  atomicAdd(&den[d * Hh + h], w);
}

// out[dst] += alpha * xl[src]  (one wave per edge, 4 channels/lane)
__global__ void edge_aggregate(const float* __restrict__ XL, const int* __restrict__ src,
                               const int* __restrict__ dst, const float* __restrict__ esc,
                               const float* __restrict__ den, float* __restrict__ out) {
  int e = (blockIdx.x * blockDim.x + threadIdx.x) >> 5;
  if (e >= Ee) return;
  int lane = threadIdx.x & 31;
  int s = src[e], d = dst[e], h = lane >> 2;
  float alpha = esc[e * Hh + h] / fmaxf(den[d * Hh + h], 1e-16f);
  float4 xl = *(const float4*)(XL + s * HC + lane * 4);
  float* o = out + d * HC + lane * 4;
  atomicAdd(o + 0, alpha * xl.x);
  atomicAdd(o + 1, alpha * xl.y);
  atomicAdd(o + 2, alpha * xl.z);
  atomicAdd(o + 3, alpha * xl.w);
}

// out = elu( (in + b - mean) * rsqrt(var+eps) * gamma + beta )
__global__ void bn_elu_k(const float* __restrict__ in, const float* __restrict__ b,
                         const float* __restrict__ g, const float* __restrict__ beta,
                         const float* __restrict__ m, const float* __restrict__ v,
                         float* __restrict__ out) {
  int t = blockIdx.x * blockDim.x + threadIdx.x;
  if (t >= Nn * HC) return;
  int c = t & (HC - 1);
  float x = in[t] + b[c];
  float y = (x - m[c]) * rsqrtf(v[c] + BN_EPS) * g[c] + beta[c];
  out[t] = y > 0.f ? y : expm1f(y);
}

// global mean pool (sums + counts); one wave per node
__global__ void pool_k(const float* __restrict__ Hd, const int* __restrict__ batch,
                       float* __restrict__ pool, float* __restrict__ cnt) {
  int n = (blockIdx.x * blockDim.x + threadIdx.x) >> 5;
  if (n >= Nn) return;
  int lane = threadIdx.x & 31;
  int g = batch[n];
  float4 hv = *(const float4*)(Hd + n * HC + lane * 4);
  float* p = pool + g * HC + lane * 4;
  atomicAdd(p + 0, hv.x);
  atomicAdd(p + 1, hv.y);
  atomicAdd(p + 2, hv.z);
  atomicAdd(p + 3, hv.w);
  if (lane == 0) atomicAdd(&cnt[g], 1.f);
}

// mean + 2-layer LSTM (1 step, h0=c0=0) + MLP head; one thread per graph
__global__ void head_k(const float* __restrict__ pool, const float* __restrict__ cnt,
                       const float* __restrict__ Wih0, const float* __restrict__ bih0,
                       const float* __restrict__ bhh0,
                       const float* __restrict__ Wih1, const float* __restrict__ bih1,
                       const float* __restrict__ bhh1,
                       const float* __restrict__ Wc1, const float* __restrict__ bc1,
                       const float* __restrict__ Wc2, const float* __restrict__ bc2,
                       float* __restrict__ out) {
  int g = threadIdx.x;
  if (g >= Gg) return;
  float inv = 1.f / fmaxf(cnt[g], 1.f);
  float gate[4 * HID];
  const float* pr = pool + g * HC;
  for (int u = 0; u < 4 * HID; ++u) {
    float acc = bih0[u] + bhh0[u];
    const float* wr = Wih0 + u * HC;
    for (int c = 0; c < HC; ++c) acc += wr[c] * (pr[c] * inv);
    gate[u] = acc;
  }
  float h1[HID];
  for (int j = 0; j < HID; ++j) {  // PyTorch gate order i,f,g,o ; f*c0 = 0
    float ci = sigmoidf(gate[j]) * tanhf(gate[2 * HID + j]);
    h1[j] = sigmoidf(gate[3 * HID + j]) * tanhf(ci);
  }
  for (int u = 0; u < 4 * HID; ++u) {
    float acc = bih1[u] + bhh1[u];
    const float* wr = Wih1 + u * HID;
    for (int c = 0; c < HID; ++c) acc += wr[c] * h1[c];
    gate[u] = acc;
  }
  float h2[HID];
  for (int j = 0; j < HID; ++j) {
    float ci = sigmoidf(gate[j]) * tanhf(gate[2 * HID + j]);
    h2[j] = sigmoidf(gate[3 * HID + j]) * tanhf(ci);
  }
  float r[HID / 2];
  for (int k = 0; k < HID / 2; ++k) {
    float acc = bc1[k];
    for (int j = 0; j < HID; ++j) acc += h2[j] * Wc1[j * (HID / 2) + k];
    r[k] = acc > 0.f ? acc : 0.f;
  }
  for (int c = 0; c < NCLS; ++c) {
    float acc = bc2[c];
    for (int k = 0; k < HID / 2; ++k) acc += r[k] * Wc2[k * NCLS + c];
    out[g * NCLS + c] = acc;
  }
}

extern "C" void kernel_launch(void* const* d_in, const int* in_sizes, int n_in,
                              void* d_out, int out_size, void* d_ws, size_t ws_size,
                              hipStream_t stream) {
  (void)in_sizes; (void)n_in; (void)out_size; (void)ws_size;
  const float* x     = (const float*)d_in[0];
  const int*   eidx  = (const int*)d_in[1];
  const int*   batch = (const int*)d_in[2];
  const int* srcp = eidx;       // edge_index[0]
  const int* dstp = eidx + Ee;  // edge_index[1]

  const float* P[35];
  for (int i = 3; i < 35; ++i) P[i] = (const float*)d_in[i];
  // layer param index bases: layer0 at 3, layer1 at 13
  // [Wl, bl, Wr, br, att, b, bn_g, bn_b, bn_m, bn_v]
  const float *Wih0 = P[23], *bih0 = P[25], *bhh0 = P[26];
  const float *Wih1 = P[27], *bih1 = P[29], *bhh1 = P[30];
  const float *Wc1 = P[31], *bc1 = P[32], *Wc2 = P[33], *bc2 = P[34];

  // ---- workspace carve-out ----
  char* ws = (char*)d_ws;
  auto carve = [&](size_t nfloats) {
    float* p = (float*)ws;
    ws += ((nfloats * sizeof(float) + 255) / 256) * 256;
    return p;
  };
  float* XL   = carve((size_t)Nn * HC);
  float* XR   = carve((size_t)Nn * HC);
  float* OUTB = carve((size_t)Nn * HC);
  float* HB   = carve((size_t)Nn * HC);
  float* ESC  = carve((size_t)Ee * Hh);
  float* EMAX = carve((size_t)Nn * Hh);
  float* DEN  = carve((size_t)Nn * Hh);
  float* POOL = carve((size_t)Gg * HC);
  float* CNT  = carve((size_t)Gg);

  const dim3 gemmGrid(Nn / 16, HC / 64);    // 3125 x 2, one wave per block
  const int edgeBlocks = (Ee * 32) / 256;   // exact: 62500
  const int ehBlocks   = (Ee * Hh) / 256;   // exact: 15625
  const int nhcBlocks  = (Nn * HC) / 256;   // exact: 25000
  const int poolBlocks = (Nn * 32) / 256;   // exact: 6250
  const int nhBlocks   = (Nn * Hh + 255) / 256;

  auto run_layer = [&](const float* Xin, int K, int base, float* Hout) {
    const float *Wl = P[base + 0], *bl = P[base + 1], *Wr = P[base + 2], *br = P[base + 3];
    const float *att = P[base + 4], *b = P[base + 5];
    const float *bg = P[base + 6], *bb = P[base + 7], *bm = P[base + 8], *bv = P[base + 9];
    if (K == FIN) {
      gemm_wmma<FIN><<<gemmGrid, 32, 0, stream>>>(Xin, Wl, bl, XL);
      gemm_wmma<FIN><<<gemmGrid, 32, 0, stream>>>(Xin, Wr, br, XR);
    } else {
      gemm_wmma<HC><<<gemmGrid, 32, 0, stream>>>(Xin, Wl, bl, XL);
      gemm_wmma<HC><<<gemmGrid, 32, 0, stream>>>(Xin, Wr, br, XR);
    }
    fill_f32<<<nhBlocks, 256, 0, stream>>>(EMAX, -INFINITY, Nn * Hh);
    hipMemsetAsync(DEN, 0, (size_t)Nn * Hh * sizeof(float), stream);
    hipMemsetAsync(OUTB, 0, (size_t)Nn * HC * sizeof(float), stream);
    edge_scores<<<edgeBlocks, 256, 0, stream>>>(XL, XR, srcp, dstp, att, ESC, EMAX);
    edge_exp<<<ehBlocks, 256, 0, stream>>>(dstp, ESC, EMAX, DEN);
    edge_aggregate<<<edgeBlocks, 256, 0, stream>>>(XL, srcp, dstp, ESC, DEN, OUTB);
    bn_elu_k<<<nhcBlocks, 256, 0, stream>>>(OUTB, b, bg, bb, bm, bv, Hout);
  };

  run_layer(x, FIN, 3, HB);   // GAT layer 0: [N,15] -> [N,128]
  run_layer(HB, HC, 13, HB);  // GAT layer 1: [N,128] -> [N,128]

  hipMemsetAsync(POOL, 0, (size_t)Gg * HC * sizeof(float), stream);
  hipMemsetAsync(CNT, 0, (size_t)Gg * sizeof(float), stream);
  pool_k<<<poolBlocks, 256, 0, stream>>>(HB, batch, POOL, CNT);
  head_k<<<1, Gg, 0, stream>>>(POOL, CNT, Wih0, bih0, bhh0, Wih1, bih1, bhh1,
                               Wc1, bc1, Wc2, bc2, (float*)d_out);
}